// GCN_3504693313815
// MI455X (gfx1250) — compile-verified
//
#include <hip/hip_runtime.h>

// GCN layer: out = relu( segment_sum(feature[src] -> dst) @ W + b )
// N=100000 nodes, E=3.2M edges, D_in = D_out = 256, all f32.
//
// Phase 1: scatter-add (atomic-bound, runs out of L2: feature 102MB + agg 102MB)
// Phase 2: GEMM + bias + relu via v_wmma_f32_16x16x4_f32 (exact f32, wave32 tiles)

typedef __attribute__((ext_vector_type(2))) float v2f;
typedef __attribute__((ext_vector_type(8))) float v8f;

#define GCN_D 256
#define GCN_N 100000

// ---------------------------------------------------------------- zero agg
__global__ void gcn_zero_ws(float4* __restrict__ p, long n4) {
    long i = (long)blockIdx.x * blockDim.x + threadIdx.x;
    if (i < n4) p[i] = make_float4(0.f, 0.f, 0.f, 0.f);
}

// ------------------------------------------------- gather + scatter-add
// 64 threads per edge; each thread moves one float4 (16B) of the 1KB row.
// Gather is coalesced (lanes read consecutive float4 of feature[src]);
// scatter is 4x global_atomic_add_f32 into agg[dst] (resolved in L2).
__global__ void gcn_scatter_add(const float* __restrict__ feature,
                                const int* __restrict__ esrc,
                                const int* __restrict__ edst,
                                float* __restrict__ agg, int E) {
    long t = (long)blockIdx.x * blockDim.x + threadIdx.x;
    int edge = (int)(t >> 6);   // 64 threads / edge
    int part = (int)(t & 63);   // float4 index within the 256-float row
    if (edge >= E) return;
    int s = esrc[edge];
    int d = edst[edge];
    const float4 v = *(const float4*)(feature + (long)s * GCN_D + part * 4);
    float* o = agg + (long)d * GCN_D + part * 4;
    atomicAdd(o + 0, v.x);
    atomicAdd(o + 1, v.y);
    atomicAdd(o + 2, v.z);
    atomicAdd(o + 3, v.w);
}

// ------------------------------------------------- GEMM + bias + relu
// One wave32 computes one 16x16 output tile, K-loop in steps of 4.
// A-matrix 16x4 f32 layout (ISA 7.12.2): lanes 0-15 hold M=lane with
// {v0=K, v1=K+1}; lanes 16-31 hold M=lane-16 with {v0=K+2, v1=K+3}.
// B-matrix 4x16: v0 = rows K / K+2 striped across lanes (N = lane&15),
// v1 = rows K+1 / K+3.  C/D: VGPR g -> (m0+g | m0+g+8, n0 + (lane&15)).
__global__ __launch_bounds__(256) void gcn_gemm_bias_relu(
    const float* __restrict__ agg, const float* __restrict__ W,
    const float* __restrict__ bias, float* __restrict__ out) {
    const int wave = threadIdx.x >> 5;
    const int lane = threadIdx.x & 31;
    const int tile = blockIdx.x * 8 + wave;   // 8 waves per block
    const int tm = tile >> 4;                 // 16 column tiles (256/16)
    const int tn = tile & 15;
    const int m0 = tm * 16, n0 = tn * 16;
    const bool hi  = lane >= 16;
    const int  l15 = lane & 15;
    const int  arow = m0 + l15;               // A row this lane carries
    const int  bcol = n0 + l15;               // B/C/D column this lane carries

    const float* aptr  = agg + (long)arow * GCN_D + (hi ? 2 : 0);
    const float* bptr0 = W + (long)(hi ? 2 : 0) * GCN_D + bcol; // v0 rows
    const float* bptr1 = W + (long)(hi ? 3 : 1) * GCN_D + bcol; // v1 rows

    v8f acc = {};
#pragma unroll 8
    for (int k = 0; k < GCN_D; k += 4) {
        // A: two consecutive K values per lane -> one aligned float2 load
        float2 af = *(const float2*)(aptr + k);
        v2f a; a.x = af.x; a.y = af.y;
        v2f b; b.x = bptr0[(long)k * GCN_D];
               b.y = bptr1[(long)k * GCN_D];
        acc = __builtin_amdgcn_wmma_f32_16x16x4_f32(
            /*neg_a=*/false, a, /*neg_b=*/false, b,
            /*c_mod=*/(short)0, acc, /*reuse_a=*/false, /*reuse_b=*/false);
    }

    const float bv = bias[bcol];
    const int rbase = m0 + (hi ? 8 : 0);
#pragma unroll
    for (int g = 0; g < 8; ++g) {
        float v = acc[g] + bv;
        out[(long)(rbase + g) * GCN_D + bcol] = (v > 0.f) ? v : 0.f;
    }
}

// ---------------------------------------------------------------- launcher
extern "C" void kernel_launch(void* const* d_in, const int* in_sizes, int n_in,
                              void* d_out, int out_size, void* d_ws, size_t ws_size,
                              hipStream_t stream) {
    const float* feature = (const float*)d_in[0];
    const int*   esrc    = (const int*)d_in[1];
    const int*   edst    = (const int*)d_in[2];
    const float* W       = (const float*)d_in[3];
    const float* bias    = (const float*)d_in[4];
    float*       out     = (float*)d_out;
    float*       agg     = (float*)d_ws;      // N * 256 f32 = 102.4 MB scratch
    const int    E       = in_sizes[1];

    // 1) zero the accumulator (harness poisons d_ws)
    long n4 = (long)GCN_N * GCN_D / 4;
    gcn_zero_ws<<<(int)((n4 + 255) / 256), 256, 0, stream>>>((float4*)agg, n4);

    // 2) gather + atomic scatter-add
    long sthreads = (long)E * 64;
    gcn_scatter_add<<<(int)((sthreads + 255) / 256), 256, 0, stream>>>(
        feature, esrc, edst, agg, E);

    // 3) WMMA GEMM + bias + relu  (100000/16 * 256/16 = 100000 tiles, 8/block)
    int ntiles = (GCN_N / 16) * (GCN_D / 16);
    gcn_gemm_bias_relu<<<ntiles / 8, 256, 0, stream>>>(agg, W, bias, out);
}